// Collision_69741678952642
// MI455X (gfx1250) — compile-verified
//
#include <hip/hip_runtime.h>

typedef __attribute__((ext_vector_type(2))) float v2f;
typedef __attribute__((ext_vector_type(8))) float v8f;

#define BB 4
#define NN 16384
#define VV 6890
#define MM 4096

// ---------------------------------------------------------------------------
// Kernel 1: gather collision subset and pack [cx, cy, cz, |c|^2] per point.
// CV layout: [B][M] float4 in d_ws (256 KB).
// ---------------------------------------------------------------------------
__global__ __launch_bounds__(256) void gather_pack_kernel(
    const float* __restrict__ collider,       // [B, V, 3]
    const int* __restrict__ cidx,             // [M]
    float4* __restrict__ cv)                  // [B, M]
{
    int t = blockIdx.x * blockDim.x + threadIdx.x;
    if (t >= BB * MM) return;
    int b = t / MM;
    int m = t - b * MM;
    int id = cidx[m];
    const float* c = collider + ((size_t)b * VV + (size_t)id) * 3;
    float x = c[0], y = c[1], z = c[2];
    cv[t] = make_float4(x, y, z, x * x + y * y + z * z);
}

// ---------------------------------------------------------------------------
// Kernel 2: block stages the full 64 KB CV[b] into LDS (all 8 waves of a
// block share one batch: 128 blocks per batch). Each wave owns one 16-vertex
// tile and runs 256 x V_WMMA_F32_16X16X4_F32 fed from conflict-free
// ds_load_b64, with a 1-deep register prefetch. S[m,n] = c.v - |c|^2/2
// = -d2/2, so argmin(d2) == argmax(S). Per-row best tracks only the tile
// index t (scalar cndmask source); m reconstructed in the epilogue.
// ---------------------------------------------------------------------------
__global__ __launch_bounds__(256) void nn_argmax_wmma_kernel(
    const float* __restrict__ vertices,       // [B, N, 3]
    const float4* __restrict__ cv,            // [B, M] packed
    int* __restrict__ out)                    // [B, N, 2]
{
    __shared__ float4 sCV[MM];                // 64 KB

    const int lane = threadIdx.x & 31;
    const int waveInBlk = threadIdx.x >> 5;
    const int b  = blockIdx.x >> 7;                        // 128 blocks/batch
    const int nt = ((blockIdx.x & 127) << 3) + waveInBlk;  // tile in batch
    const int n0 = nt << 4;
    const int col = lane & 15;
    const bool hi = lane >= 16;

    // Stage CV[b] into LDS: 256 threads x 16 float4 (b128 in, b128 out).
    {
        const float4* cvb = cv + (size_t)b * MM;
#pragma unroll
        for (int i = 0; i < MM / 256; ++i)
            sCV[i * 256 + threadIdx.x] = cvb[i * 256 + threadIdx.x];
    }

    // B operand (vertices): loaded once per wave. 12-byte stride -> scalar
    // loads (only 4-byte alignment guaranteed).
    const float* vtx = vertices + ((size_t)b * NN + (size_t)(n0 + col)) * 3;
    v2f bmat;
    if (!hi) { bmat.x = vtx[0]; bmat.y = vtx[1]; }   // K0=vx, K1=vy
    else     { bmat.x = vtx[2]; bmat.y = -0.5f; }    // K2=vz, K3=-0.5

    __syncthreads();

    // Lane's LDS read slot: (tile*16 + col) float4s, +8B for the high half.
    const float* scvf = (const float*)sCV;
    const int laneOfs = col * 4 + (hi ? 2 : 0);

    float bestv[8];
    int   bestt[8];
#pragma unroll
    for (int r = 0; r < 8; ++r) { bestv[r] = -3.402823466e38f; bestt[r] = 0; }

    float2 cur = *(const float2*)(scvf + laneOfs);   // tile 0

    for (int t = 0; t < MM / 16; ++t) {
        // Prefetch next tile's A fragment (wraps harmlessly on last iter).
        float2 nxt = *(const float2*)(scvf + (((t + 1) & 255) << 6) + laneOfs);

        v2f amat; amat.x = cur.x; amat.y = cur.y;
        v8f acc = {};
        acc = __builtin_amdgcn_wmma_f32_16x16x4_f32(
            /*neg_a=*/false, amat, /*neg_b=*/false, bmat,
            /*c_mod=*/(short)0, acc, /*reuse_a=*/false, /*reuse_b=*/false);

#pragma unroll
        for (int r = 0; r < 8; ++r) {
            bool gt = acc[r] > bestv[r];             // ties keep earlier t
            bestv[r] = gt ? acc[r] : bestv[r];
            bestt[r] = gt ? t      : bestt[r];
        }
        cur = nxt;
    }

    // Epilogue: reconstruct m and reduce 8 rows branchlessly (low m wins ties).
    const int rowBase = hi ? 8 : 0;
    float bv = bestv[0];
    int   bi = bestt[0] * 16 + rowBase;
#pragma unroll
    for (int r = 1; r < 8; ++r) {
        int   mi = bestt[r] * 16 + rowBase + r;
        bool  better = (bestv[r] > bv) | ((bestv[r] == bv) & (mi < bi));
        bv = better ? bestv[r] : bv;
        bi = better ? mi       : bi;
    }
    // Merge the two lane halves (same vertex n, complementary m rows).
    {
        float ov = __shfl_xor(bv, 16, 32);
        int   oi = __shfl_xor(bi, 16, 32);
        bool  better = (ov > bv) | ((ov == bv) & (oi < bi));
        bv = better ? ov : bv;
        bi = better ? oi : bi;
    }

    if (!hi) {
        int2* o = (int2*)out;
        o[(size_t)b * NN + (size_t)(n0 + col)] = make_int2(b, bi);
    }
}

// ---------------------------------------------------------------------------
// Host launcher. Inputs: vertices f32 [B,N,3], collider f32 [B,V,3],
// collision_vertices i32 [M]. Output: i32 [B,N,2]. Scratch: 256 KB for CV.
// ---------------------------------------------------------------------------
extern "C" void kernel_launch(void* const* d_in, const int* in_sizes, int n_in,
                              void* d_out, int out_size, void* d_ws, size_t ws_size,
                              hipStream_t stream) {
    const float* vertices = (const float*)d_in[0];
    const float* collider = (const float*)d_in[1];
    const int*   cidx     = (const int*)d_in[2];
    int*         out      = (int*)d_out;
    float4*      cv       = (float4*)d_ws;    // BB*MM*16 bytes = 256 KB

    (void)in_sizes; (void)n_in; (void)out_size; (void)ws_size;

    // Pack gathered collider points with their squared norms.
    {
        int total = BB * MM;
        gather_pack_kernel<<<(total + 255) / 256, 256, 0, stream>>>(
            collider, cidx, cv);
    }
    // One wave per 16-vertex tile: BB * (NN/16) = 4096 waves = 512 blocks.
    {
        int blocks = BB * (NN / 16) / 8;      // 8 waves (256 threads) / block
        nn_argmax_wmma_kernel<<<blocks, 256, 0, stream>>>(
            vertices, cv, out);
    }
}